// GATContextSubgraphClassifier_26731876451135
// MI455X (gfx1250) — compile-verified
//
#include <hip/hip_runtime.h>
#include <hip/hip_bf16.h>
#include <math.h>

typedef __attribute__((ext_vector_type(2))) float v2f;
typedef __attribute__((ext_vector_type(8))) float v8f;

#define N_NODES 50000
#define N_EDGES 800000
#define F_IN    128
#define HID     64
#define HEADS   4
#define G_GRAPH 512
#define NEGSLOPE 0.2f
#define LN_EPS   1e-5f
#define STAT_BLOCKS 1024

// ---------------------------------------------------------------------------
// Generic fills / copies
// ---------------------------------------------------------------------------
__global__ __launch_bounds__(256) void fill_int_kernel(int* p, int v, long n) {
    long i = (long)blockIdx.x * blockDim.x + threadIdx.x;
    long stride = (long)gridDim.x * blockDim.x;
    for (; i < n; i += stride) p[i] = v;
}

__global__ __launch_bounds__(256) void copy_int_kernel(const int* a, int* b, long n) {
    long i = (long)blockIdx.x * blockDim.x + threadIdx.x;
    if (i < n) b[i] = a[i];
}

// ---------------------------------------------------------------------------
// GEMM  out[M,Nout] = A[M,K] @ W[K,Nout] + bias, optional ReLU.
// One wave32 per 16x16 output tile, V_WMMA_F32_16X16X4_F32 over K.
// Requires M%16==0, K%4==0, Nout%16==0 (true for all call sites).
// ---------------------------------------------------------------------------
__global__ __launch_bounds__(256) void gemm_bias_wmma(
    const float* __restrict__ A, const float* __restrict__ W,
    const float* __restrict__ bias, float* __restrict__ out,
    int M, int K, int Nout, int fuse_relu)
{
    int gtid  = blockIdx.x * blockDim.x + threadIdx.x;
    int wave  = gtid >> 5;                 // uniform across the wave
    int lane  = threadIdx.x & 31;
    int tilesN = Nout >> 4;
    int tilesM = M >> 4;
    if (wave >= tilesM * tilesN) return;   // whole-wave uniform exit (EXEC stays all-1s)
    int tileM = wave / tilesN;
    int tileN = wave - tileM * tilesN;

    int half = lane >> 4;                  // 0: lanes 0-15, 1: lanes 16-31
    int ln   = lane & 15;
    int row  = tileM * 16 + ln;            // A-matrix M index for this lane
    int col  = tileN * 16 + ln;            // B/D N index for this lane

    const float* arow = A + (size_t)row * K;
    v8f acc = {};
    for (int k0 = 0; k0 < K; k0 += 4) {
        int ka = k0 + 2 * half;
        v2f a, b;
        a.x = arow[ka];
        a.y = arow[ka + 1];
        b.x = W[(size_t)ka * Nout + col];
        b.y = W[(size_t)(ka + 1) * Nout + col];
        acc = __builtin_amdgcn_wmma_f32_16x16x4_f32(
            /*neg_a=*/false, a, /*neg_b=*/false, b,
            /*c_mod=*/(short)0, acc, /*reuse_a=*/false, /*reuse_b=*/false);
    }

    float bv = bias ? bias[col] : 0.0f;
#pragma unroll
    for (int v = 0; v < 8; ++v) {
        int r = tileM * 16 + v + 8 * half; // D layout: VGPR v holds M=v (+8 for upper lanes)
        float val = acc[v] + bv;
        if (fuse_relu) val = fmaxf(val, 0.0f);
        out[(size_t)r * Nout + col] = val;
    }
}

// ---------------------------------------------------------------------------
// CSR build over dst (edge list is layer-invariant; built once per call)
// ---------------------------------------------------------------------------
__global__ __launch_bounds__(256) void deg_kernel(const int* __restrict__ dst,
                                                  int* __restrict__ deg) {
    int e = blockIdx.x * blockDim.x + threadIdx.x;
    if (e < N_EDGES) atomicAdd(&deg[dst[e]], 1);
}

__global__ __launch_bounds__(1024) void scan_kernel(const int* __restrict__ deg,
                                                    int* __restrict__ rowptr) {
    __shared__ int buf[1024];
    __shared__ int carry;
    int tid = threadIdx.x;
    if (tid == 0) { carry = 0; rowptr[0] = 0; }
    __syncthreads();
    for (int base = 0; base < N_NODES; base += 1024) {
        int v = (base + tid < N_NODES) ? deg[base + tid] : 0;
        buf[tid] = v;
        __syncthreads();
        for (int o = 1; o < 1024; o <<= 1) {          // Hillis-Steele inclusive scan
            int t = (tid >= o) ? buf[tid - o] : 0;
            __syncthreads();
            buf[tid] += t;
            __syncthreads();
        }
        if (base + tid < N_NODES) rowptr[base + tid + 1] = carry + buf[tid];
        __syncthreads();
        if (tid == 0) carry += buf[1023];
        __syncthreads();
    }
}

__global__ __launch_bounds__(256) void bucket_kernel(const int* __restrict__ dst,
                                                     int* __restrict__ cursor,
                                                     int* __restrict__ eidx) {
    int e = blockIdx.x * blockDim.x + threadIdx.x;
    if (e >= N_EDGES) return;
    int p = atomicAdd(&cursor[dst[e]], 1);
    eidx[p] = e;
}

// Sort each node's segment by edge id -> deterministic accumulation order.
__global__ __launch_bounds__(256) void sortseg_kernel(const int* __restrict__ rowptr,
                                                      int* __restrict__ eidx) {
    int n = blockIdx.x * blockDim.x + threadIdx.x;
    if (n >= N_NODES) return;
    int beg = rowptr[n], end = rowptr[n + 1];
    for (int i = beg + 1; i < end; ++i) {
        int v = eidx[i];
        int j = i - 1;
        while (j >= beg && eidx[j] > v) { eidx[j + 1] = eidx[j]; --j; }
        eidx[j + 1] = v;
    }
}

// ---------------------------------------------------------------------------
// Fused GATv2 edge stage (gather form, no atomics).
// One wave32 per (node, head). Lane owns channels {lane, lane+32} (HID=64).
// Pass 1: per incident edge, cooperative 64-dot -> logit (wave shfl reduce),
//         stash logit in CSR-slot order, track running max.
// Pass 2: w = exp(logit - max); accumulate denominator and weighted xl[src];
//         normalize at the end (softmax division folded into epilogue).
// ---------------------------------------------------------------------------
__global__ __launch_bounds__(256) void node_agg_kernel(
    const float* __restrict__ xl, const float* __restrict__ xr,
    const float* __restrict__ att,
    const int* __restrict__ rowptr, const int* __restrict__ eidx,
    const int* __restrict__ src,
    float* __restrict__ elog, float* __restrict__ out, int H)
{
    int gtid = blockIdx.x * blockDim.x + threadIdx.x;
    int wave = gtid >> 5;                  // uniform across the wave
    int lane = threadIdx.x & 31;
    if (wave >= N_NODES * H) return;
    int n = wave / H;
    int h = wave - n * H;

    int beg = rowptr[n], end = rowptr[n + 1];
    const size_t nb = (size_t)n * H * HID + (size_t)h * HID;
    float xr0 = xr[nb + lane];
    float xr1 = xr[nb + lane + 32];
    float at0 = att[h * HID + lane];
    float at1 = att[h * HID + lane + 32];

    // ---- pass 1: logits + segment max ----
    float m = -3.402823466e38f;
    for (int i = beg; i < end; ++i) {
        int e = eidx[i];
        const float* xp = xl + (size_t)src[e] * H * HID + (size_t)h * HID;
        if (i + 1 < end)
            __builtin_prefetch(xl + (size_t)src[eidx[i + 1]] * H * HID + (size_t)h * HID, 0, 3);
        float v0 = xp[lane] + xr0;        v0 = (v0 > 0.0f) ? v0 : v0 * NEGSLOPE;
        float v1 = xp[lane + 32] + xr1;   v1 = (v1 > 0.0f) ? v1 : v1 * NEGSLOPE;
        float p = v0 * at0 + v1 * at1;
#pragma unroll
        for (int o = 16; o > 0; o >>= 1) p += __shfl_xor(p, o, 32);
        if (lane == 0) elog[(size_t)i * H + h] = p;   // CSR-slot indexed: conflict-free
        m = fmaxf(m, p);                               // p uniform after reduction
    }

    // ---- pass 2: softmax weights + weighted aggregation ----
    float den = 0.0f, a0 = 0.0f, a1 = 0.0f;
    for (int i = beg; i < end; ++i) {
        int e = eidx[i];
        float w = __expf(elog[(size_t)i * H + h] - m);  // uniform broadcast read
        den += w;
        const float* xp = xl + (size_t)src[e] * H * HID + (size_t)h * HID;
        a0 += w * xp[lane];
        a1 += w * xp[lane + 32];
    }
    float inv = (den > 0.0f) ? 1.0f / den : 0.0f;       // empty segment -> 0 (matches segment_sum)
    out[nb + lane]      = a0 * inv;
    out[nb + lane + 32] = a1 * inv;
}

// ---------------------------------------------------------------------------
// Bias add (in place) + per-block partial sum/sumsq (deterministic two-pass LN)
// ---------------------------------------------------------------------------
__global__ __launch_bounds__(256) void bias_stats_kernel(
    float* __restrict__ h, const float* __restrict__ bias,
    float* __restrict__ part, long total, int D)
{
    __shared__ float s0[256];
    __shared__ float s1[256];
    long i = (long)blockIdx.x * blockDim.x + threadIdx.x;
    long stride = (long)gridDim.x * blockDim.x;
    float s = 0.0f, q = 0.0f;
    for (; i < total; i += stride) {
        float val = h[i] + bias[(int)(i % D)];
        h[i] = val;
        s += val;
        q += val * val;
    }
    s0[threadIdx.x] = s;
    s1[threadIdx.x] = q;
    __syncthreads();
    for (int o = 128; o > 0; o >>= 1) {
        if ((int)threadIdx.x < o) {
            s0[threadIdx.x] += s0[threadIdx.x + o];
            s1[threadIdx.x] += s1[threadIdx.x + o];
        }
        __syncthreads();
    }
    if (threadIdx.x == 0) {
        part[2 * blockIdx.x]     = s0[0];
        part[2 * blockIdx.x + 1] = s1[0];
    }
}

__global__ void finalize_stats_kernel(const float* part, float* st, float cnt) {
    float s = 0.0f, q = 0.0f;
    for (int i = 0; i < STAT_BLOCKS; ++i) { s += part[2 * i]; q += part[2 * i + 1]; }
    float mu  = s / cnt;
    float var = q / cnt - mu * mu;
    st[0] = mu;
    st[1] = rsqrtf(fmaxf(var, 0.0f) + LN_EPS);
}

__global__ __launch_bounds__(256) void ln_relu_res_kernel(
    float* __restrict__ h, const float* __restrict__ g, const float* __restrict__ b,
    const float* __restrict__ st, const float* __restrict__ res, long total, int D)
{
    long i = (long)blockIdx.x * blockDim.x + threadIdx.x;
    if (i >= total) return;
    float mu = st[0], rs = st[1];
    int col = (int)(i % D);
    float v = (h[i] - mu) * rs * g[col] + b[col];
    v = fmaxf(v, 0.0f);
    if (res) v += res[i];
    h[i] = v;
}

// ---------------------------------------------------------------------------
// Global mean pool: batch is sorted -> binary-search the node range per graph.
// One thread per (graph, feature); deterministic, no atomics.
// ---------------------------------------------------------------------------
__global__ __launch_bounds__(256) void pool_kernel(
    const float* __restrict__ h, const int* __restrict__ batch, float* __restrict__ Z)
{
    int idx = blockIdx.x * blockDim.x + threadIdx.x;
    if (idx >= G_GRAPH * HID) return;
    int g = idx >> 6;
    int f = idx & 63;
    int lo = 0, hi = N_NODES;
    while (lo < hi) { int mid = (lo + hi) >> 1; if (batch[mid] < g) lo = mid + 1; else hi = mid; }
    int beg = lo;
    lo = 0; hi = N_NODES;
    while (lo < hi) { int mid = (lo + hi) >> 1; if (batch[mid] < g + 1) lo = mid + 1; else hi = mid; }
    int end = lo;
    float s = 0.0f;
    for (int n = beg; n < end; ++n) s += h[(size_t)n * HID + f];
    Z[idx] = s / fmaxf((float)(end - beg), 1.0f);
}

// ---------------------------------------------------------------------------
// Final projection HID -> 1 (one thread per graph)
// ---------------------------------------------------------------------------
__global__ __launch_bounds__(256) void mlp2_kernel(
    const float* __restrict__ ZR, const float* __restrict__ Wh2,
    const float* __restrict__ bh2, float* __restrict__ out)
{
    int g = blockIdx.x * blockDim.x + threadIdx.x;
    if (g >= G_GRAPH) return;
    float a = 0.0f;
#pragma unroll 8
    for (int k = 0; k < HID; ++k) a += ZR[(size_t)g * HID + k] * Wh2[k];
    out[g] = a + bh2[0];
}

// ---------------------------------------------------------------------------
// Host-side driver
// ---------------------------------------------------------------------------
static inline int cdiv_i(long a, long b) { return (int)((a + b - 1) / b); }

static void run_gat_layer(const float* hin, int din, int H,
                          const float* Wl, const float* bl,
                          const float* Wr, const float* br,
                          const float* att, const float* bias,
                          const float* ln_g, const float* ln_b,
                          const int* rowptr, const int* eidx, const int* src,
                          float* XL, float* XR, float* ELOG, float* PART, float* ST,
                          float* hout, bool residual, hipStream_t stream)
{
    const int D = H * HID;
    // xl = hin @ Wl + bl ; xr = hin @ Wr + br   (WMMA GEMMs)
    int waves = (N_NODES / 16) * (D / 16);
    int grid  = cdiv_i(waves, 8);
    gemm_bias_wmma<<<grid, 256, 0, stream>>>(hin, Wl, bl, XL, N_NODES, din, D, 0);
    gemm_bias_wmma<<<grid, 256, 0, stream>>>(hin, Wr, br, XR, N_NODES, din, D, 0);

    // fused edge stage (gather): logits + softmax + aggregation, no atomics
    long aggthreads = (long)N_NODES * H * 32;
    node_agg_kernel<<<cdiv_i(aggthreads, 256), 256, 0, stream>>>(
        XL, XR, att, rowptr, eidx, src, ELOG, hout, H);

    // bias + graph-layernorm stats (deterministic), then LN + ReLU (+ residual)
    long total = (long)N_NODES * D;
    bias_stats_kernel<<<STAT_BLOCKS, 256, 0, stream>>>(hout, bias, PART, total, D);
    finalize_stats_kernel<<<1, 1, 0, stream>>>(PART, ST, (float)total);
    ln_relu_res_kernel<<<cdiv_i(total, 256), 256, 0, stream>>>(
        hout, ln_g, ln_b, ST, residual ? hin : nullptr, total, D);
}

extern "C" void kernel_launch(void* const* d_in, const int* in_sizes, int n_in,
                              void* d_out, int out_size, void* d_ws, size_t ws_size,
                              hipStream_t stream)
{
    (void)in_sizes; (void)n_in; (void)out_size; (void)ws_size;

    const float* x     = (const float*)d_in[0];
    const int*   ei    = (const int*)d_in[1];
    const int*   batch = (const int*)d_in[2];
    const int*   src   = ei;
    const int*   dst   = ei + N_EDGES;

    // per-layer params: base 3 + 8*l : Wl, bl, Wr, br, att, bias, ln_g, ln_b
    const float* P[24];
    for (int i = 0; i < 24; ++i) P[i] = (const float*)d_in[3 + i];
    const float* Wh1 = (const float*)d_in[27];
    const float* bh1 = (const float*)d_in[28];
    const float* Wh2 = (const float*)d_in[29];
    const float* bh2 = (const float*)d_in[30];
    float* out = (float*)d_out;

    // workspace carve-up: float region, then int region
    float* ws = (float*)d_ws;
    size_t off = 0;
    float* XL   = ws + off; off += (size_t)N_NODES * 256;
    float* XR   = ws + off; off += (size_t)N_NODES * 256;
    float* H1   = ws + off; off += (size_t)N_NODES * 256;
    float* H2   = ws + off; off += (size_t)N_NODES * HID;
    float* H3   = ws + off; off += (size_t)N_NODES * HID;
    float* ELOG = ws + off; off += (size_t)N_EDGES * HEADS;
    float* Z    = ws + off; off += (size_t)G_GRAPH * HID;
    float* ZR   = ws + off; off += (size_t)G_GRAPH * HID;
    float* PART = ws + off; off += 2 * STAT_BLOCKS;
    float* ST   = ws + off; off += 8;
    int* iws    = (int*)(ws + off);
    size_t ioff = 0;
    int* DEG    = iws + ioff; ioff += N_NODES;
    int* ROWPTR = iws + ioff; ioff += N_NODES + 1;
    int* CURSOR = iws + ioff; ioff += N_NODES;
    int* EIDX   = iws + ioff; ioff += N_EDGES;

    // ---- build dst-CSR once (shared by all three layers) ----
    fill_int_kernel<<<256, 256, 0, stream>>>(DEG, 0, N_NODES);
    deg_kernel<<<cdiv_i(N_EDGES, 256), 256, 0, stream>>>(dst, DEG);
    scan_kernel<<<1, 1024, 0, stream>>>(DEG, ROWPTR);
    copy_int_kernel<<<cdiv_i(N_NODES, 256), 256, 0, stream>>>(ROWPTR, CURSOR, N_NODES);
    bucket_kernel<<<cdiv_i(N_EDGES, 256), 256, 0, stream>>>(dst, CURSOR, EIDX);
    sortseg_kernel<<<cdiv_i(N_NODES, 256), 256, 0, stream>>>(ROWPTR, EIDX);

    // ---- three GATv2 layers ----
    run_gat_layer(x,  F_IN,      HEADS, P[0],  P[1],  P[2],  P[3],  P[4],  P[5],  P[6],  P[7],
                  ROWPTR, EIDX, src, XL, XR, ELOG, PART, ST, H1, false, stream);
    run_gat_layer(H1, HID*HEADS, 1,     P[8],  P[9],  P[10], P[11], P[12], P[13], P[14], P[15],
                  ROWPTR, EIDX, src, XL, XR, ELOG, PART, ST, H2, false, stream);
    run_gat_layer(H2, HID,       1,     P[16], P[17], P[18], P[19], P[20], P[21], P[22], P[23],
                  ROWPTR, EIDX, src, XL, XR, ELOG, PART, ST, H3, true,  stream);

    // ---- global mean pool (binary search on sorted batch; no atomics) ----
    pool_kernel<<<cdiv_i(G_GRAPH * HID, 256), 256, 0, stream>>>(H3, batch, Z);

    // ---- MLP head: relu(Z @ Wh1 + bh1) @ Wh2 + bh2 ----
    {
        int waves = (G_GRAPH / 16) * (HID / 16);
        gemm_bias_wmma<<<cdiv_i(waves, 8), 256, 0, stream>>>(Z, Wh1, bh1, ZR, G_GRAPH, HID, HID, 1);
    }
    mlp2_kernel<<<cdiv_i(G_GRAPH, 256), 256, 0, stream>>>(ZR, Wh2, bh2, out);
}